// HardNegativePositivePointLoss_45251775431302
// MI455X (gfx1250) — compile-verified
//
#include <hip/hip_runtime.h>

// ---------------------------------------------------------------------------
// HardNegativePositivePointLoss for MI455X (gfx1250)
// points [256,256] f32, memory_bank [262144,256] f32 (rows L2-normalized),
// point_indices [256] i32, first (0) -> outputs (loss1, loss2) as 2 floats.
//
// Pipeline (deterministic; integer/fixed-point atomics only):
//   k_norm   : normalize points, fold in 1/T
//   k_gemm   : scores = pnorm @ mb^T via V_WMMA_F32_16X16X4_F32 (LDS-staged)
//   k_hist   : per-row 2048-bin histogram (counts u32 + exp-sums u64 fixed pt)
//              -> boundary bins + exact exp-sums of bins above, for k=4096/51
//   k_select : collect boundary-bin elements to LDS, exact 21-bit radix select
//              -> exact thresholds, tie counts, final sums, per-row losses
//   k_reduce : mean over rows -> d_out[0..1]
// ---------------------------------------------------------------------------

#define B_ROWS 256
#define N_MEM  262144
#define DIM    256
#define TOPK   4096
#define TOPS   51          // K+1 in the reference
#define INV_T  (1.0f / 0.07f)

#define GEMM_NB   64       // memory entries per workgroup
#define LDS_ROW   260      // padded row stride (floats): 16B aligned, bank-safe

#define LIST_CAP  16384    // boundary-bin list capacity (expected ~4-5K)
#define FIX_SCALE 1.099511627776e12      // 2^40
#define FIX_INV   9.094947017729282e-13  // 2^-40

typedef __attribute__((ext_vector_type(2))) float v2f;
typedef __attribute__((ext_vector_type(4))) float v4f;
typedef __attribute__((ext_vector_type(8))) float v8f;

// monotone float <-> uint encoding (larger float -> larger uint)
__device__ __forceinline__ unsigned encf(float f) {
    unsigned u = __float_as_uint(f);
    return u ^ ((u >> 31) ? 0xFFFFFFFFu : 0x80000000u);
}
__device__ __forceinline__ float decf(unsigned u) {
    unsigned x = (u & 0x80000000u) ? (u ^ 0x80000000u) : ~u;
    return __uint_as_float(x);
}
__device__ __forceinline__ unsigned long long fixpt(float e) {
    return (unsigned long long)((double)e * FIX_SCALE);
}

// ---------------------------------------------------------------------------
// 1) normalize points, fold in 1/T:  pnorm = p / (||p|| * T)
// ---------------------------------------------------------------------------
__global__ void k_norm(const float* __restrict__ pts, float* __restrict__ pnorm) {
    __shared__ float red[256];
    const int row = blockIdx.x, t = threadIdx.x;
    float v = pts[row * DIM + t];
    red[t] = v * v;
    __syncthreads();
    for (int s = 128; s > 0; s >>= 1) {
        if (t < s) red[t] += red[t + s];
        __syncthreads();
    }
    float inv = INV_T / sqrtf(red[0]);
    pnorm[row * DIM + t] = v * inv;
}

// ---------------------------------------------------------------------------
// 2) GEMM: scores[b][n] = pnorm[b,:] . mb[n,:]   via V_WMMA_F32_16X16X4_F32
//    WG = 256 threads (8 waves). WG owns 64 memory rows (LDS tile) x all 256
//    points. Wave w owns M-tiles {2w,2w+1} x 4 N-tiles of 16.
// ---------------------------------------------------------------------------
__global__ void k_gemm(const float* __restrict__ pnorm,
                       const float* __restrict__ mb,
                       float* __restrict__ scores) {
    extern __shared__ float smb[];                 // [GEMM_NB][LDS_ROW]
    const int tid = threadIdx.x;
    const long long n0 = (long long)blockIdx.x * GEMM_NB;

    // stage 64x256 bank tile into LDS (coalesced float4 loads)
    const v4f* mb4 = (const v4f*)(mb + n0 * DIM);
    #pragma unroll
    for (int i = 0; i < 16; ++i) {
        int idx = tid + i * 256;                   // 0..4095 float4 slots
        int r   = idx >> 6;                        // row (64 float4 per row)
        int c4  = idx & 63;
        v4f v = mb4[(long long)r * 64 + c4];
        float* dst = &smb[r * LDS_ROW + c4 * 4];
        dst[0] = v[0]; dst[1] = v[1]; dst[2] = v[2]; dst[3] = v[3];
    }
    __syncthreads();

    const int wave = tid >> 5, lane = tid & 31;
    const int m_base = wave * 32;
    const int lhalf = lane & 15;
    const int khi   = (lane >> 4) * 2;             // K sub-offset per A/B layout

    v8f acc[2][4] = {};

    const float* aBase0 = pnorm + (size_t)(m_base + lhalf) * DIM + khi;
    const float* aBase1 = aBase0 + 16 * DIM;
    const float* bBase  = smb + (size_t)lhalf * LDS_ROW + khi;

    for (int k0 = 0; k0 < DIM; k0 += 4) {
        v2f a0 = *(const v2f*)(aBase0 + k0);
        v2f a1 = *(const v2f*)(aBase1 + k0);
        v2f b0 = *(const v2f*)(bBase + 0 * 16 * LDS_ROW + k0);
        v2f b1 = *(const v2f*)(bBase + 1 * 16 * LDS_ROW + k0);
        v2f b2 = *(const v2f*)(bBase + 2 * 16 * LDS_ROW + k0);
        v2f b3 = *(const v2f*)(bBase + 3 * 16 * LDS_ROW + k0);
        acc[0][0] = __builtin_amdgcn_wmma_f32_16x16x4_f32(false, a0, false, b0, (short)0, acc[0][0], false, false);
        acc[0][1] = __builtin_amdgcn_wmma_f32_16x16x4_f32(false, a0, false, b1, (short)0, acc[0][1], false, false);
        acc[0][2] = __builtin_amdgcn_wmma_f32_16x16x4_f32(false, a0, false, b2, (short)0, acc[0][2], false, false);
        acc[0][3] = __builtin_amdgcn_wmma_f32_16x16x4_f32(false, a0, false, b3, (short)0, acc[0][3], false, false);
        acc[1][0] = __builtin_amdgcn_wmma_f32_16x16x4_f32(false, a1, false, b0, (short)0, acc[1][0], false, false);
        acc[1][1] = __builtin_amdgcn_wmma_f32_16x16x4_f32(false, a1, false, b1, (short)0, acc[1][1], false, false);
        acc[1][2] = __builtin_amdgcn_wmma_f32_16x16x4_f32(false, a1, false, b2, (short)0, acc[1][2], false, false);
        acc[1][3] = __builtin_amdgcn_wmma_f32_16x16x4_f32(false, a1, false, b3, (short)0, acc[1][3], false, false);
    }

    // C/D layout: VGPR r -> lanes 0-15: M=r, lanes 16-31: M=r+8; N = lane&15
    #pragma unroll
    for (int mt = 0; mt < 2; ++mt) {
        #pragma unroll
        for (int nt = 0; nt < 4; ++nt) {
            size_t rowb = (size_t)(m_base + mt * 16 + ((lane >> 4) * 8));
            size_t col  = (size_t)n0 + nt * 16 + lhalf;
            #pragma unroll
            for (int r = 0; r < 8; ++r)
                scores[(rowb + r) * (size_t)N_MEM + col] = acc[mt][nt][r];
        }
    }
}

// ---------------------------------------------------------------------------
// 3) pass 1: per-row 2048-bin histogram over bits 31..21 of the monotone
//    encoding; u32 counts + u64 fixed-point exp-sums (order-independent).
//    stateU per row: [0]=binA [1]=remA [2]=binB [3]=remB
//    stateS per row: [0]=sum_above_A  [1]=sum_above_B   (u64 fixed point)
// ---------------------------------------------------------------------------
__global__ void k_hist(const float* __restrict__ scores,
                       unsigned* __restrict__ stateU,
                       unsigned long long* __restrict__ stateS) {
    __shared__ unsigned cnt[2048];
    __shared__ unsigned long long sum[2048];
    const int row = blockIdx.x, t = threadIdx.x;
    for (int i = t; i < 2048; i += 256) { cnt[i] = 0u; sum[i] = 0ull; }
    __syncthreads();
    const float* srow = scores + (size_t)row * N_MEM;
    for (int n = t; n < N_MEM; n += 256) {
        float v = srow[n];
        unsigned u = encf(v);
        unsigned b = u >> 21;
        atomicAdd(&cnt[b], 1u);
        atomicAdd(&sum[b], fixpt(expf(v)));
    }
    __syncthreads();
    if (t == 0) {
        unsigned cum = 0; unsigned long long csum = 0ull;
        int ba = -1, bb = -1; unsigned ra = 0, rb = 0;
        unsigned long long sa = 0ull, sb = 0ull;
        for (int bin = 2047; bin >= 0; --bin) {
            unsigned c = cnt[bin];
            if (bb < 0 && cum + c >= TOPS) { bb = bin; rb = TOPS - cum; sb = csum; }
            if (ba < 0 && cum + c >= TOPK) { ba = bin; ra = TOPK - cum; sa = csum; }
            cum += c; csum += sum[bin];
            if (ba >= 0) break;
        }
        unsigned* st = stateU + row * 4;
        st[0] = (unsigned)ba; st[1] = ra; st[2] = (unsigned)bb; st[3] = rb;
        stateS[row * 2 + 0] = sa;
        stateS[row * 2 + 1] = sb;
    }
}

// bit-serial radix select over 21-bit list values: find value of the k-th
// largest element; returns prefix (threshold low bits), *remOut = ties to take
__device__ unsigned radix21(const unsigned* L, int n, unsigned k, int t,
                            unsigned* cshare, unsigned* remOut) {
    unsigned prefix = 0, remK = k;
    for (int bit = 20; bit >= 0; --bit) {
        if (t == 0) *cshare = 0u;
        __syncthreads();
        unsigned kmask = (0xFFFFFFFFu << (bit + 1)) & 0x1FFFFFu;
        unsigned bitm  = 1u << bit;
        unsigned local = 0;
        for (int i = t; i < n; i += 256) {
            unsigned x = L[i];
            if ((x & kmask) == prefix && (x & bitm)) local++;
        }
        atomicAdd(cshare, local);
        __syncthreads();
        unsigned c = *cshare;
        if (c >= remK) prefix |= bitm; else remK -= c;
        __syncthreads();
    }
    *remOut = remK;
    return prefix;
}

// order-independent fixed-point sum of exp over list elements > prefix
__device__ unsigned long long sumgt(const unsigned* L, int n, unsigned prefix,
                                    unsigned hibin, int t,
                                    unsigned long long* sshare) {
    if (t == 0) *sshare = 0ull;
    __syncthreads();
    unsigned long long local = 0ull;
    for (int i = t; i < n; i += 256) {
        unsigned x = L[i];
        if (x > prefix) local += fixpt(expf(decf((hibin << 21) | x)));
    }
    atomicAdd(sshare, local);
    __syncthreads();
    return *sshare;
}

// ---------------------------------------------------------------------------
// 4) pass 2: collect boundary-bin elements into LDS lists, resolve exact
//    thresholds + tie counts, finish denom / top-51 sums, per-row losses.
// ---------------------------------------------------------------------------
__global__ void k_select(const float* __restrict__ scores,
                         const int* __restrict__ pidx,
                         const unsigned* __restrict__ stateU,
                         const unsigned long long* __restrict__ stateS,
                         float* __restrict__ rowloss) {
    extern __shared__ unsigned lists[];            // La[LIST_CAP], Lb[LIST_CAP]
    __shared__ unsigned cntA, cntB, cshare;
    __shared__ unsigned long long sshare;
    unsigned* La = lists;
    unsigned* Lb = lists + LIST_CAP;

    const int row = blockIdx.x, t = threadIdx.x;
    const unsigned* st = stateU + row * 4;
    const unsigned b1a = st[0], k1a = st[1];
    const unsigned b1b = st[2], k1b = st[3];
    if (t == 0) { cntA = 0u; cntB = 0u; }
    __syncthreads();

    const float* srow = scores + (size_t)row * N_MEM;
    for (int n = t; n < N_MEM; n += 256) {
        unsigned u = encf(srow[n]);
        unsigned hi = u >> 21;
        if (hi == b1a) { unsigned i = atomicAdd(&cntA, 1u); if (i < LIST_CAP) La[i] = u & 0x1FFFFFu; }
        if (hi == b1b) { unsigned i = atomicAdd(&cntB, 1u); if (i < LIST_CAP) Lb[i] = u & 0x1FFFFFu; }
    }
    __syncthreads();
    const int nA = (int)min(cntA, (unsigned)LIST_CAP);
    const int nB = (int)min(cntB, (unsigned)LIST_CAP);

    unsigned remA, remB;
    unsigned pA = radix21(La, nA, k1a, t, &cshare, &remA);
    unsigned long long gA = sumgt(La, nA, pA, b1a, t, &sshare);
    __syncthreads();
    unsigned pB = radix21(Lb, nB, k1b, t, &cshare, &remB);
    unsigned long long gB = sumgt(Lb, nB, pB, b1b, t, &sshare);

    if (t == 0) {
        double denom = (double)(stateS[row * 2 + 0] + gA) * FIX_INV
                     + (double)remA * (double)expf(decf((b1a << 21) | pA));
        double top51 = (double)(stateS[row * 2 + 1] + gB) * FIX_INV
                     + (double)remB * (double)expf(decf((b1b << 21) | pB));
        int idx = pidx[row];
        float spos = srow[idx];
        float ps = expf(spos);
        unsigned utB = (b1b << 21) | pB;
        float hasSelf = (encf(spos) >= utB) ? 1.0f : 0.0f;
        double allOther = top51 - (double)(hasSelf * ps);
        rowloss[row]          = -logf((float)((double)ps / denom) + 1e-7f);
        rowloss[B_ROWS + row] = -logf((float)(allOther / denom));
    }
}

__global__ void k_reduce(const float* __restrict__ rowloss, float* __restrict__ out) {
    __shared__ float r1[256], r2[256];
    const int t = threadIdx.x;
    r1[t] = rowloss[t];
    r2[t] = rowloss[B_ROWS + t];
    __syncthreads();
    for (int s = 128; s > 0; s >>= 1) {
        if (t < s) { r1[t] += r1[t + s]; r2[t] += r2[t + s]; }
        __syncthreads();
    }
    if (t == 0) {
        out[0] = r1[0] / (float)B_ROWS;
        out[1] = r2[0] / (float)B_ROWS;
    }
}

// ---------------------------------------------------------------------------
extern "C" void kernel_launch(void* const* d_in, const int* in_sizes, int n_in,
                              void* d_out, int out_size, void* d_ws, size_t ws_size,
                              hipStream_t stream) {
    const float* points = (const float*)d_in[0];
    const float* mb     = (const float*)d_in[1];
    const int*   pidx   = (const int*)d_in[2];
    float*       out    = (float*)d_out;

    char* ws = (char*)d_ws;
    const size_t sc_b = (size_t)B_ROWS * N_MEM * sizeof(float);       // 256 MB
    float*              pnorm   = (float*)ws;                         // 256 KB
    float*              scores  = (float*)(ws + (size_t)(1 << 20));
    unsigned*           stateU  = (unsigned*)(ws + (size_t)(1 << 20) + sc_b);
    unsigned long long* stateS  = (unsigned long long*)(ws + (size_t)(1 << 20) + sc_b + 8192);
    float*              rowloss = (float*)(ws + (size_t)(1 << 20) + sc_b + 32768);

    k_norm<<<B_ROWS, 256, 0, stream>>>(points, pnorm);
    k_gemm<<<N_MEM / GEMM_NB, 256, GEMM_NB * LDS_ROW * sizeof(float), stream>>>(pnorm, mb, scores);
    k_hist<<<B_ROWS, 256, 0, stream>>>(scores, stateU, stateS);
    k_select<<<B_ROWS, 256, 2 * LIST_CAP * sizeof(unsigned), stream>>>(scores, pidx, stateU, stateS, rowloss);
    k_reduce<<<1, 256, 0, stream>>>(rowloss, out);
    (void)in_sizes; (void)n_in; (void)out_size; (void)ws_size;
}